// Decodeur_3899830305148
// MI455X (gfx1250) — compile-verified
//
#include <hip/hip_runtime.h>
#include <hip/hip_bf16.h>
#include <math.h>

// Problem constants (fixed by reference)
#define PAD 0
#define EOS 1
#define SOS 2
#define VOCAB 32000
#define EDIM 256
#define HDIM 512
#define GDIM (3 * HDIM)
#define BATCH 32
#define TLEN 64

// proj partials: 250 waves, each covering 8 n-tiles of 16 cols = 128 cols
#define PROJ_WAVES 250
#define NT_PER_WAVE 8

typedef float v2f __attribute__((ext_vector_type(2)));
typedef float v8f __attribute__((ext_vector_type(8)));

// ---------------------------------------------------------------------------
// WMMA f32 16x16x4 fragment loader.
// A-matrix 16x4 (MxK): lanes 0-15 -> M=lane, VGPR0=K0,VGPR1=K1; lanes 16-31 -> K2,K3.
// B-matrix 4x16 (KxN) built from row-major W[N,K] (B = W^T): same addressing with
// row = n-base + (lane&15).  So one helper serves both.
// ---------------------------------------------------------------------------
__device__ __forceinline__ v2f frag2(const float* M, int rowBase, int ld, int k) {
    int lane = threadIdx.x & 31;
    int r = rowBase + (lane & 15);
    int kk = k + ((lane >> 4) << 1);
    const float* p = M + (size_t)r * ld + kk;
    return v2f{p[0], p[1]};
}

__device__ __forceinline__ v8f wmma_f32(v2f a, v2f b, v8f c) {
    return __builtin_amdgcn_wmma_f32_16x16x4_f32(
        /*neg_a=*/false, a, /*neg_b=*/false, b,
        /*c_mod=*/(short)0, c, /*reuse_a=*/false, /*reuse_b=*/false);
}

// ---------------------------------------------------------------------------
// init: h = hidden[0], x = emb[SOS], done = 0
// ---------------------------------------------------------------------------
__global__ void init_kernel(const float* __restrict__ hidden,
                            const float* __restrict__ emb,
                            float* __restrict__ h, float* __restrict__ x,
                            int* __restrict__ done) {
    int i = blockIdx.x * blockDim.x + threadIdx.x;
    if (i < BATCH * HDIM) h[i] = hidden[i];
    if (i < BATCH * EDIM) x[i] = emb[(size_t)SOS * EDIM + (i & (EDIM - 1))];
    if (i < BATCH) done[i] = 0;
}

// ---------------------------------------------------------------------------
// gates: gx = x @ W_ih^T + b_ih  (jobs 0..95, K=EDIM)
//        gh = h @ W_hh^T + b_hh  (jobs 96..191, K=HDIM)
// one wave (32 threads) per 16-column N-tile, both 16-row M-tiles.
// ---------------------------------------------------------------------------
__global__ void gates_kernel(const float* __restrict__ x, const float* __restrict__ h,
                             const float* __restrict__ W_ih, const float* __restrict__ W_hh,
                             const float* __restrict__ b_ih, const float* __restrict__ b_hh,
                             float* __restrict__ gx, float* __restrict__ gh) {
    int lane = threadIdx.x;
    int job = blockIdx.x;
    bool is_gx = job < (GDIM / 16);
    int n0 = (is_gx ? job : job - (GDIM / 16)) * 16;
    const float* A = is_gx ? x : h;
    const float* W = is_gx ? W_ih : W_hh;
    const float* bias = is_gx ? b_ih : b_hh;
    float* out = is_gx ? gx : gh;
    int K = is_gx ? EDIM : HDIM;

    v8f c0 = {}, c1 = {};
    for (int k = 0; k < K; k += 4) {
        v2f a0 = frag2(A, 0, K, k);
        v2f a1 = frag2(A, 16, K, k);
        v2f bf = frag2(W, n0, K, k);
        c0 = wmma_f32(a0, bf, c0);
        c1 = wmma_f32(a1, bf, c1);
    }
    int col = n0 + (lane & 15);
    float bb = bias[col];
    int half8 = (lane >> 4) << 3;  // +8 rows for lanes 16-31
#pragma unroll
    for (int r = 0; r < 8; ++r) {
        int row = r + half8;
        out[(size_t)row * GDIM + col] = c0[r] + bb;
        out[(size_t)(row + 16) * GDIM + col] = c1[r] + bb;
    }
}

// ---------------------------------------------------------------------------
// GRU elementwise update, h in place (gh already computed from old h)
// ---------------------------------------------------------------------------
__device__ __forceinline__ float sigmoidf_(float v) { return 1.0f / (1.0f + expf(-v)); }

__global__ void gru_update_kernel(const float* __restrict__ gx, const float* __restrict__ gh,
                                  float* __restrict__ h) {
    int i = blockIdx.x * blockDim.x + threadIdx.x;
    if (i >= BATCH * HDIM) return;
    int b = i >> 9;          // /HDIM
    int j = i & (HDIM - 1);
    const float* gxr = gx + (size_t)b * GDIM;
    const float* ghr = gh + (size_t)b * GDIM;
    float r = sigmoidf_(gxr[j] + ghr[j]);
    float z = sigmoidf_(gxr[HDIM + j] + ghr[HDIM + j]);
    float n = tanhf(gxr[2 * HDIM + j] + r * ghr[2 * HDIM + j]);
    h[i] = (1.0f - z) * n + z * h[i];
}

// ---------------------------------------------------------------------------
// projection + partial argmax.
// grid = 250 single-wave blocks; each wave covers 8 n-tiles (128 vocab cols),
// both M-tiles, then reduces argmax per row via lane shuffles (wave32 halves).
// ---------------------------------------------------------------------------
__global__ void proj_kernel(const float* __restrict__ h, const float* __restrict__ W_proj,
                            const float* __restrict__ b_proj,
                            float* __restrict__ pval, int* __restrict__ pidx) {
    int lane = threadIdx.x;
    int wid = blockIdx.x;

    float bestv[2][8];
    int besti[2][8];
#pragma unroll
    for (int m = 0; m < 2; ++m)
#pragma unroll
        for (int r = 0; r < 8; ++r) { bestv[m][r] = -INFINITY; besti[m][r] = 0x7fffffff; }

    for (int nt = 0; nt < NT_PER_WAVE; ++nt) {
        int n0 = (wid * NT_PER_WAVE + nt) * 16;
        v8f c0 = {}, c1 = {};
        for (int k = 0; k < HDIM; k += 8) {
            v2f a0 = frag2(h, 0, HDIM, k);
            v2f a1 = frag2(h, 16, HDIM, k);
            v2f bf = frag2(W_proj, n0, HDIM, k);
            c0 = wmma_f32(a0, bf, c0);
            c1 = wmma_f32(a1, bf, c1);
            v2f a0b = frag2(h, 0, HDIM, k + 4);
            v2f a1b = frag2(h, 16, HDIM, k + 4);
            v2f bfb = frag2(W_proj, n0, HDIM, k + 4);
            c0 = wmma_f32(a0b, bfb, c0);
            c1 = wmma_f32(a1b, bfb, c1);
        }
        int col = n0 + (lane & 15);
        float bb = b_proj[col];
#pragma unroll
        for (int m = 0; m < 2; ++m) {
#pragma unroll
            for (int r = 0; r < 8; ++r) {
                float v = (m ? c1[r] : c0[r]) + bb;
                int idx = col;
                // reduce over the 16 lanes of this half (masks 1,2,4,8 stay in-half)
#pragma unroll
                for (int s = 1; s < 16; s <<= 1) {
                    float ov = __shfl_xor(v, s, 32);
                    int oi = __shfl_xor(idx, s, 32);
                    if (ov > v || (ov == v && oi < idx)) { v = ov; idx = oi; }
                }
                if (v > bestv[m][r] || (v == bestv[m][r] && idx < besti[m][r])) {
                    bestv[m][r] = v; besti[m][r] = idx;
                }
            }
        }
    }
    // lanes 0 / 16 hold results for rows r and r+8 respectively
    if ((lane & 15) == 0) {
        int half8 = (lane >> 4) << 3;
#pragma unroll
        for (int m = 0; m < 2; ++m)
#pragma unroll
            for (int r = 0; r < 8; ++r) {
                int row = m * 16 + r + half8;
                pval[(size_t)row * PROJ_WAVES + wid] = bestv[m][r];
                pidx[(size_t)row * PROJ_WAVES + wid] = besti[m][r];
            }
    }
}

// ---------------------------------------------------------------------------
// finalize: reduce partial argmax, apply done/EOS logic, emit token (as float),
// gather next-step embedding x = emb[tok]. one block per batch row.
// ---------------------------------------------------------------------------
__global__ void finalize_kernel(const float* __restrict__ pval, const int* __restrict__ pidx,
                                int* __restrict__ done, const float* __restrict__ emb,
                                float* __restrict__ x, float* __restrict__ out_t) {
    int b = blockIdx.x;
    int tid = threadIdx.x;
    __shared__ float sv[256];
    __shared__ int si[256];
    __shared__ int stok;

    float bv = -INFINITY;
    int bi = 0x7fffffff;
    for (int j = tid; j < PROJ_WAVES; j += 256) {
        float v = pval[(size_t)b * PROJ_WAVES + j];
        int i = pidx[(size_t)b * PROJ_WAVES + j];
        if (v > bv || (v == bv && i < bi)) { bv = v; bi = i; }
    }
    sv[tid] = bv; si[tid] = bi;
    __syncthreads();
    for (int s = 128; s > 0; s >>= 1) {
        if (tid < s) {
            float ov = sv[tid + s]; int oi = si[tid + s];
            if (ov > sv[tid] || (ov == sv[tid] && oi < si[tid])) { sv[tid] = ov; si[tid] = oi; }
        }
        __syncthreads();
    }
    if (tid == 0) {
        int y_new = si[0];
        int d = done[b];
        int tok = d ? EOS : y_new;
        done[b] = d | (y_new == EOS);
        out_t[b] = (float)tok;
        stok = tok;
    }
    __syncthreads();
    int tok = stok;
    for (int j = tid; j < EDIM; j += 256)
        x[(size_t)b * EDIM + j] = emb[(size_t)tok * EDIM + j];
}

// ---------------------------------------------------------------------------
extern "C" void kernel_launch(void* const* d_in, const int* in_sizes, int n_in,
                              void* d_out, int out_size, void* d_ws, size_t ws_size,
                              hipStream_t stream) {
    const float* hidden = (const float*)d_in[0];
    const float* emb    = (const float*)d_in[1];
    const float* W_ih   = (const float*)d_in[2];
    const float* W_hh   = (const float*)d_in[3];
    const float* b_ih   = (const float*)d_in[4];
    const float* b_hh   = (const float*)d_in[5];
    const float* W_proj = (const float*)d_in[6];
    const float* b_proj = (const float*)d_in[7];
    float* out = (float*)d_out;

    float* ws = (float*)d_ws;
    float* h    = ws;                           // BATCH*HDIM   = 16384
    float* x    = h + BATCH * HDIM;             // BATCH*EDIM   = 8192
    float* gx   = x + BATCH * EDIM;             // BATCH*GDIM   = 49152
    float* gh   = gx + BATCH * GDIM;            // BATCH*GDIM   = 49152
    float* pval = gh + BATCH * GDIM;            // BATCH*250    = 8000
    int*   pidx = (int*)(pval + BATCH * PROJ_WAVES);
    int*   done = pidx + BATCH * PROJ_WAVES;    // BATCH

    init_kernel<<<(BATCH * HDIM + 255) / 256, 256, 0, stream>>>(hidden, emb, h, x, done);

    for (int t = 0; t < TLEN; ++t) {
        gates_kernel<<<2 * (GDIM / 16), 32, 0, stream>>>(x, h, W_ih, W_hh, b_ih, b_hh, gx, gh);
        gru_update_kernel<<<(BATCH * HDIM + 255) / 256, 256, 0, stream>>>(gx, gh, h);
        proj_kernel<<<PROJ_WAVES, 32, 0, stream>>>(h, W_proj, b_proj, pval, pidx);
        finalize_kernel<<<BATCH, 256, 0, stream>>>(pval, pidx, done, emb, x,
                                                   out + (size_t)t * BATCH);
    }
}